// Embedding_58935541235998
// MI455X (gfx1250) — compile-verified
//
#include <hip/hip_runtime.h>

typedef __attribute__((ext_vector_type(2))) float v2f;
typedef __attribute__((ext_vector_type(4))) float v4f;
typedef __attribute__((ext_vector_type(8))) float v8f;

#define NB   128
#define DIN  128
#define DOUT 64
#define TOK_PER_WG 128          // 8 waves * 16 tokens each
#define XPAD 132                // DIN + 4 dwords: row stride ≡ 4 (mod 64 banks)
#define SWZW_FLOATS (16 * 4 * 32 * 4)   // k2 x ntile x lane x 4 = 8192 floats (32KB)
#define LDS_FLOATS (SWZW_FLOATS + TOK_PER_WG * XPAD)

// Block-diagonal linear via v_wmma_f32_16x16x4_f32 (exact f32 math).
// W is staged into LDS in *fragment order*: one ds_load_b128 per lane yields
// the B pairs for two consecutive K-steps, pre-positioned in even VGPR quads.
__global__ __launch_bounds__(256) void blockdiag_linear_wmma(
    const float* __restrict__ x,     // [tokens, NB*DIN]
    const float* __restrict__ W,     // [NB, DIN, DOUT]
    const float* __restrict__ bias,  // [NB, DOUT]
    float* __restrict__ y)           // [tokens, NB*DOUT]
{
    extern __shared__ float lds[];
    float* swzW = lds;                    // fragment-ordered W: 8192 floats
    float* ldsX = lds + SWZW_FLOATS;      // [TOK_PER_WG][XPAD]

    const int n    = blockIdx.x;                  // block 0..127
    const int twg  = blockIdx.y * TOK_PER_WG;     // first token of this WG
    const int tid  = threadIdx.x;
    const int lane = tid & 31;
    const int wave = tid >> 5;

    // ---- stage W[n] into LDS in WMMA B-fragment order ----
    // slot idx -> (k2, t, lane'): lane' holds col = t*16+(lane'&15),
    // rows {8k2+ko, 8k2+ko+1, 8k2+4+ko, 8k2+5+ko}, ko = (lane'>>4)*2.
    {
        const float* Wg = W + (size_t)n * (DIN * DOUT);
        #pragma unroll
        for (int i = 0; i < (SWZW_FLOATS / 4) / 256; ++i) {   // 8 iters
            const int idx = tid + i * 256;                    // b128 slot id
            const int lp  = idx & 31;
            const int s   = idx >> 5;                         // 0..63
            const int k2  = s >> 2;
            const int t   = s & 3;
            const int c   = t * 16 + (lp & 15);
            const int r   = 8 * k2 + ((lp >> 4) << 1);
            v4f v;
            v[0] = Wg[(r    ) * DOUT + c];
            v[1] = Wg[(r + 1) * DOUT + c];
            v[2] = Wg[(r + 4) * DOUT + c];
            v[3] = Wg[(r + 5) * DOUT + c];
            *(v4f*)(swzW + idx * 4) = v;
        }
    }
    // ---- stage x tile (128 tokens x 128 feats); 512B contiguous per wave ----
    {
        const float* Xg = x + (size_t)twg * (NB * DIN) + (size_t)n * DIN;
        #pragma unroll
        for (int i = 0; i < (TOK_PER_WG * DIN / 4) / 256; ++i) {  // 16 iters
            const int idx = tid + i * 256;
            const int row = idx >> 5;                       // 32 float4 / row
            const int c4  = idx & 31;
            const v4f v   = *(const v4f*)(Xg + (size_t)row * (NB * DIN) + c4 * 4);
            *(v4f*)(ldsX + row * XPAD + c4 * 4) = v;
        }
    }
    __syncthreads();

    // ---- per-wave 16x64 output tile: 4 accumulators of 16x16 ----
    const int mrow = lane & 15;             // A row / B col / D col
    const int koff = (lane >> 4) << 1;      // K pair select: 0 or 2
    const float* xrow  = ldsX + (wave * 16 + mrow) * XPAD;
    const float* wbase = swzW + lane * 4;

    v8f acc[4];
    #pragma unroll
    for (int t = 0; t < 4; ++t) {
        const float bv = bias[n * DOUT + t * 16 + mrow];
        #pragma unroll
        for (int r = 0; r < 8; ++r) acc[t][r] = bv;
    }

    #pragma unroll 4
    for (int k2 = 0; k2 < DIN / 8; ++k2) {    // two K-steps per iteration
        // A frags: merged by compiler into one ds_load_2addr_b64
        const v2f a0 = *(const v2f*)(xrow + 8 * k2     + koff);
        const v2f a1 = *(const v2f*)(xrow + 8 * k2 + 4 + koff);
        #pragma unroll
        for (int t = 0; t < 4; ++t) {
            const v4f q = *(const v4f*)(wbase + (size_t)(k2 * 4 + t) * 128);
            const v2f b0 = __builtin_shufflevector(q, q, 0, 1);
            const v2f b1 = __builtin_shufflevector(q, q, 2, 3);
            acc[t] = __builtin_amdgcn_wmma_f32_16x16x4_f32(
                false, a0, false, b0, (short)0, acc[t], false, false);
            acc[t] = __builtin_amdgcn_wmma_f32_16x16x4_f32(
                false, a1, false, b1, (short)0, acc[t], false, false);
        }
    }

    // ---- store: C/D VGPR r holds rows {r, r+8} split across lane halves ----
    float* yg = y + (size_t)(twg + wave * 16) * (NB * DOUT) + (size_t)n * DOUT;
    const int rbase = (lane >> 4) << 3;     // 0 or 8
    #pragma unroll
    for (int t = 0; t < 4; ++t) {
        #pragma unroll
        for (int r = 0; r < 8; ++r) {
            yg[(size_t)(rbase + r) * (NB * DOUT) + t * 16 + mrow] = acc[t][r];
        }
    }
}

extern "C" void kernel_launch(void* const* d_in, const int* in_sizes, int n_in,
                              void* d_out, int out_size, void* d_ws, size_t ws_size,
                              hipStream_t stream) {
    const float* x    = (const float*)d_in[0];   // [B*S, NB*DIN] f32
    const float* W    = (const float*)d_in[1];   // [NB, DIN, DOUT] f32
    const float* bias = (const float*)d_in[2];   // [NB, DOUT] f32
    float* y = (float*)d_out;

    const int tokens = in_sizes[0] / (NB * DIN); // 4096
    const size_t lds_bytes = (size_t)LDS_FLOATS * sizeof(float); // ~98 KB (<320 KB/WGP)

    // Allow >64KB dynamic LDS on CDNA5 (320KB per WGP).
    (void)hipFuncSetAttribute((const void*)blockdiag_linear_wmma,
                              hipFuncAttributeMaxDynamicSharedMemorySize,
                              (int)lds_bytes);

    dim3 grid(NB, tokens / TOK_PER_WG, 1);       // (128, 32)
    dim3 block(256, 1, 1);                       // 8 waves (wave32)
    blockdiag_linear_wmma<<<grid, block, lds_bytes, stream>>>(x, W, bias, y);
}